// GATNodeClassifier_47115791237314
// MI455X (gfx1250) — compile-verified
//
#include <hip/hip_runtime.h>
#include <hip/hip_bf16.h>

// ---------------------------------------------------------------------------
// GAT node classifier, fp32, CDNA5 (gfx1250, wave32).
//  Pipeline per layer:  WMMA gemm (+alpha)  ->  init accum  ->  edge max ->
//                       edge exp/sum        ->  edge aggregate (atomics) ->
//                       bias+ELU.  Then a small fused MLP head.
// ---------------------------------------------------------------------------

typedef __attribute__((ext_vector_type(2))) float v2f;
typedef __attribute__((ext_vector_type(8))) float v8f;

#define NEG_SLOPE 0.2f

// order-preserving float <-> uint mapping for atomic max on signed floats
__device__ __forceinline__ unsigned int f2u_order(float f) {
    unsigned int b = __float_as_uint(f);
    return (b & 0x80000000u) ? ~b : (b | 0x80000000u);
}
__device__ __forceinline__ float u2f_order(unsigned int u) {
    unsigned int b = (u & 0x80000000u) ? (u & 0x7FFFFFFFu) : ~u;
    return __uint_as_float(b);
}

// ---------------------------------------------------------------------------
// K1: h = X @ W  ([N,Fin] x [Fin,64]) with fused attention coefficients
//     alpha_src[n,h] = sum_c h[n,h,c]*a_src[h,c]   (same for alpha_dst).
// One workgroup = 16 nodes x 64 cols; 4 waves, one 16x16 WMMA tile per wave
// (wave index == head index since 64/16 == H == 4).  K-loop of fp32 WMMA.
// ---------------------------------------------------------------------------
__global__ __launch_bounds__(128)
void gat_gemm_alpha(const float* __restrict__ X, const float* __restrict__ W,
                    const float* __restrict__ a_src, const float* __restrict__ a_dst,
                    float* __restrict__ Hout,
                    float* __restrict__ alpha_s, float* __restrict__ alpha_d,
                    int N, int Fin)
{
    __shared__ float xs[16 * 128];      // X tile [16][Fin], Fin <= 128
    __shared__ float hs[4][16][17];     // per-wave 16x16 result tile (padded)

    const int tid  = threadIdx.x;
    const int wave = tid >> 5;          // 0..3 : column block == head
    const int lane = tid & 31;
    const int half = lane >> 4;         // lane group (ISA 16x16 layouts)
    const int id   = lane & 15;
    const int node0 = blockIdx.x * 16;  // N is a multiple of 16 here (50000)

    // cooperative float4 load of the 16xFin X tile (rows are contiguous)
    {
        const int n4 = (16 * Fin) >> 2;
        const float4* Xv = (const float4*)(X + (size_t)node0 * Fin);
        float4* xsv = (float4*)xs;
        for (int i = tid; i < n4; i += 128) xsv[i] = Xv[i];
    }
    __syncthreads();

    v8f c = {};
    const int colBase = wave * 16;
    for (int k0 = 0; k0 < Fin; k0 += 4) {
        const int k = k0 + 2 * half;
        v2f a, b;
        // A 16x4 f32: lanes 0-15 -> K=0,1 ; lanes 16-31 -> K=2,3 (ISA 7.12.2)
        a.x = xs[id * Fin + k];
        a.y = xs[id * Fin + k + 1];
        // B 4x16 f32 (transpose-symmetric layout)
        b.x = W[(size_t)k * 64 + colBase + id];
        b.y = W[(size_t)(k + 1) * 64 + colBase + id];
        c = __builtin_amdgcn_wmma_f32_16x16x4_f32(false, a, false, b,
                                                  (short)0, c, false, false);
    }

    // C/D layout: lanes 0-15 -> M=r, lanes 16-31 -> M=8+r, N=id
    #pragma unroll
    for (int r = 0; r < 8; ++r) {
        const int mrow = r + 8 * half;
        const float v = c[r];
        hs[wave][mrow][id] = v;
        Hout[(size_t)(node0 + mrow) * 64 + colBase + id] = v;
    }
    __syncthreads();

    // lanes 0-15: alpha_src of node id; lanes 16-31: alpha_dst of node id
    {
        const float* avec = (half == 0) ? a_src : a_dst;
        float s = 0.f;
        #pragma unroll
        for (int n = 0; n < 16; ++n) s += hs[wave][id][n] * avec[wave * 16 + n];
        float* dp = (half == 0) ? alpha_s : alpha_d;
        dp[(size_t)(node0 + id) * 4 + wave] = s;
    }
}

// ---------------------------------------------------------------------------
// K2: reset accumulators (out=0, den=0, m_u=0 which decodes to -inf-ish; every
// node has a self-loop so every m/den gets at least one contribution).
// ---------------------------------------------------------------------------
__global__ void init_accum(float* __restrict__ out, unsigned int* __restrict__ m_u,
                           float* __restrict__ den, int N)
{
    const int i = blockIdx.x * blockDim.x + threadIdx.x;
    if (i < N * 64) out[i] = 0.f;
    if (i < N * 4) { m_u[i] = 0u; den[i] = 0.f; }
}

// ---------------------------------------------------------------------------
// K3: segment max of leaky-relu(e) per (dst, head), via u32 atomic max.
// One thread per edge handles all 4 heads (alpha rows are exactly float4).
// ---------------------------------------------------------------------------
__global__ void gat_edge_max(const long long* __restrict__ srcp,
                             const long long* __restrict__ dstp,
                             int E, int Etot,
                             const float* __restrict__ alpha_s,
                             const float* __restrict__ alpha_d,
                             unsigned int* __restrict__ m_u)
{
    const int e = blockIdx.x * blockDim.x + threadIdx.x;
    if (e >= Etot) return;
    long long s, d;
    if (e < E) { s = srcp[e]; d = dstp[e]; } else { s = e - E; d = s; }
    const float4 as = *(const float4*)(alpha_s + s * 4);
    const float4 ad = *(const float4*)(alpha_d + d * 4);
    const float ee[4] = {as.x + ad.x, as.y + ad.y, as.z + ad.z, as.w + ad.w};
    #pragma unroll
    for (int h = 0; h < 4; ++h) {
        const float v = ee[h] > 0.f ? ee[h] : NEG_SLOPE * ee[h];
        atomicMax(&m_u[d * 4 + h], f2u_order(v));
    }
}

// ---------------------------------------------------------------------------
// K4: ex = exp(e - m[dst]); den[dst] += ex (atomic); stash ex for pass C.
// ---------------------------------------------------------------------------
__global__ void gat_edge_expsum(const long long* __restrict__ srcp,
                                const long long* __restrict__ dstp,
                                int E, int Etot,
                                const float* __restrict__ alpha_s,
                                const float* __restrict__ alpha_d,
                                const unsigned int* __restrict__ m_u,
                                float* __restrict__ exbuf,
                                float* __restrict__ den)
{
    const int e = blockIdx.x * blockDim.x + threadIdx.x;
    if (e >= Etot) return;
    if (e + 8192 < E) __builtin_prefetch(&srcp[e + 8192], 0, 1);  // global_prefetch_b8
    long long s, d;
    if (e < E) { s = srcp[e]; d = dstp[e]; } else { s = e - E; d = s; }
    const float4 as = *(const float4*)(alpha_s + s * 4);
    const float4 ad = *(const float4*)(alpha_d + d * 4);
    const uint4  mu = *(const uint4*)(m_u + d * 4);
    const float ee[4] = {as.x + ad.x, as.y + ad.y, as.z + ad.z, as.w + ad.w};
    const unsigned int mm[4] = {mu.x, mu.y, mu.z, mu.w};
    float4 exv;
    float* exp_out = (float*)&exv;
    #pragma unroll
    for (int h = 0; h < 4; ++h) {
        float v = ee[h] > 0.f ? ee[h] : NEG_SLOPE * ee[h];
        const float x = __expf(v - u2f_order(mm[h]));
        exp_out[h] = x;
        atomicAdd(&den[d * 4 + h], x);
    }
    *(float4*)(exbuf + (size_t)e * 4) = exv;
}

// ---------------------------------------------------------------------------
// K5: out[dst] += h[src] * (ex/den[dst]).  16 threads per edge: lane j owns
// channels [4j,4j+4) -> the 256B h[src] row is read coalesced as float4s.
// ---------------------------------------------------------------------------
__global__ void gat_edge_aggregate(const long long* __restrict__ srcp,
                                   const long long* __restrict__ dstp,
                                   int E, int Etot,
                                   const float* __restrict__ Hmat,
                                   const float* __restrict__ exbuf,
                                   const float* __restrict__ den,
                                   float* __restrict__ out)
{
    const long long t = (long long)blockIdx.x * blockDim.x + threadIdx.x;
    if (t >= (long long)Etot * 16) return;
    const int e = (int)(t >> 4);
    const int j = (int)(t & 15);
    const int head = j >> 2;
    long long s, d;
    if (e < E) { s = srcp[e]; d = dstp[e]; } else { s = e - E; d = s; }
    const float att = exbuf[(size_t)e * 4 + head] / den[d * 4 + head];
    const float4 hv = *(const float4*)(Hmat + s * 64 + j * 4);
    float* o = out + d * 64 + j * 4;
    atomicAdd(o + 0, hv.x * att);
    atomicAdd(o + 1, hv.y * att);
    atomicAdd(o + 2, hv.z * att);
    atomicAdd(o + 3, hv.w * att);
}

// ---------------------------------------------------------------------------
// K6: out = elu(out + bias)
// ---------------------------------------------------------------------------
__global__ void bias_elu(float* __restrict__ out, const float* __restrict__ b, int N)
{
    const int i = blockIdx.x * blockDim.x + threadIdx.x;
    if (i >= N * 64) return;
    const float v = out[i] + b[i & 63];
    out[i] = v > 0.f ? v : (__expf(v) - 1.f);
}

// ---------------------------------------------------------------------------
// K7: y = relu(h @ W3 + b3) @ W4 + b4   (64 -> 32 -> 2), W3/W4 staged in LDS.
// ---------------------------------------------------------------------------
__global__ __launch_bounds__(256)
void final_mlp(const float* __restrict__ Hin,
               const float* __restrict__ W3, const float* __restrict__ b3,
               const float* __restrict__ W4, const float* __restrict__ b4,
               float* __restrict__ out, int N)
{
    __shared__ float w3s[64 * 32];
    __shared__ float w4s[64];
    __shared__ float b3s[32];
    __shared__ float b4s[2];
    for (int i = threadIdx.x; i < 64 * 32; i += 256) w3s[i] = W3[i];
    if (threadIdx.x < 64) w4s[threadIdx.x] = W4[threadIdx.x];
    if (threadIdx.x < 32) b3s[threadIdx.x] = b3[threadIdx.x];
    if (threadIdx.x < 2)  b4s[threadIdx.x] = b4[threadIdx.x];
    __syncthreads();

    const int n = blockIdx.x * 256 + threadIdx.x;
    if (n >= N) return;
    const float* hr = Hin + (size_t)n * 64;
    float hreg[64];
    #pragma unroll
    for (int k = 0; k < 64; ++k) hreg[k] = hr[k];
    float y0 = b4s[0], y1 = b4s[1];
    for (int j = 0; j < 32; ++j) {
        float sacc = b3s[j];
        #pragma unroll
        for (int k = 0; k < 64; ++k) sacc += hreg[k] * w3s[k * 32 + j];
        sacc = sacc > 0.f ? sacc : 0.f;
        y0 += sacc * w4s[j * 2 + 0];
        y1 += sacc * w4s[j * 2 + 1];
    }
    out[(size_t)n * 2 + 0] = y0;
    out[(size_t)n * 2 + 1] = y1;
}

// ---------------------------------------------------------------------------
extern "C" void kernel_launch(void* const* d_in, const int* in_sizes, int n_in,
                              void* d_out, int out_size, void* d_ws, size_t ws_size,
                              hipStream_t stream)
{
    const float*     x    = (const float*)d_in[0];
    const long long* ei   = (const long long*)d_in[1];   // int64 [2,E]
    const float*     W1   = (const float*)d_in[2];
    const float*     as1  = (const float*)d_in[3];
    const float*     ad1  = (const float*)d_in[4];
    const float*     b1   = (const float*)d_in[5];
    const float*     W2   = (const float*)d_in[6];
    const float*     as2  = (const float*)d_in[7];
    const float*     ad2  = (const float*)d_in[8];
    const float*     b2   = (const float*)d_in[9];
    const float*     W3   = (const float*)d_in[10];
    const float*     b3   = (const float*)d_in[11];
    const float*     W4   = (const float*)d_in[12];
    const float*     b4   = (const float*)d_in[13];

    const int FIN  = 128, HID = 64;
    const int N    = in_sizes[0] / FIN;     // 50000
    const int E    = in_sizes[1] / 2;       // 1600000
    const int Etot = E + N;                 // + self loops
    const long long* srcp = ei;
    const long long* dstp = ei + E;

    // workspace carve-up (floats), all 16B aligned
    float* ws = (float*)d_ws;
    size_t off = 0;
    float*        h_buf   = ws + off; off += (size_t)N * 64;
    float*        out_buf = ws + off; off += (size_t)N * 64;
    float*        alpha_s = ws + off; off += (size_t)N * 4;
    float*        alpha_d = ws + off; off += (size_t)N * 4;
    unsigned int* m_u     = (unsigned int*)(ws + off); off += (size_t)N * 4;
    float*        den     = ws + off; off += (size_t)N * 4;
    float*        exbuf   = ws + off; off += (size_t)Etot * 4;

    const int gGemm  = (N + 15) / 16;
    const int gInit  = (N * 64 + 255) / 256;
    const int gEdge  = (Etot + 255) / 256;
    const int gAgg   = (int)(((long long)Etot * 16 + 255) / 256);
    const int gMlp   = (N + 255) / 256;

    // -------- layer 1 --------
    gat_gemm_alpha<<<gGemm, 128, 0, stream>>>(x, W1, as1, ad1, h_buf, alpha_s, alpha_d, N, FIN);
    init_accum<<<gInit, 256, 0, stream>>>(out_buf, m_u, den, N);
    gat_edge_max<<<gEdge, 256, 0, stream>>>(srcp, dstp, E, Etot, alpha_s, alpha_d, m_u);
    gat_edge_expsum<<<gEdge, 256, 0, stream>>>(srcp, dstp, E, Etot, alpha_s, alpha_d, m_u, exbuf, den);
    gat_edge_aggregate<<<gAgg, 256, 0, stream>>>(srcp, dstp, E, Etot, h_buf, exbuf, den, out_buf);
    bias_elu<<<gInit, 256, 0, stream>>>(out_buf, b1, N);

    // -------- layer 2 (gemm reads out_buf BEFORE it is re-zeroed) --------
    gat_gemm_alpha<<<gGemm, 128, 0, stream>>>(out_buf, W2, as2, ad2, h_buf, alpha_s, alpha_d, N, HID);
    init_accum<<<gInit, 256, 0, stream>>>(out_buf, m_u, den, N);
    gat_edge_max<<<gEdge, 256, 0, stream>>>(srcp, dstp, E, Etot, alpha_s, alpha_d, m_u);
    gat_edge_expsum<<<gEdge, 256, 0, stream>>>(srcp, dstp, E, Etot, alpha_s, alpha_d, m_u, exbuf, den);
    gat_edge_aggregate<<<gAgg, 256, 0, stream>>>(srcp, dstp, E, Etot, h_buf, exbuf, den, out_buf);
    bias_elu<<<gInit, 256, 0, stream>>>(out_buf, b2, N);

    // -------- MLP head --------
    final_mlp<<<gMlp, 256, 0, stream>>>(out_buf, W3, b3, W4, b4, (float*)d_out, N);
}